// LambdaLayer_89172110999766
// MI455X (gfx1250) — compile-verified
//
#include <hip/hip_runtime.h>
#include <math.h>

typedef __attribute__((ext_vector_type(2))) float v2f;
typedef __attribute__((ext_vector_type(8))) float v8f;

#define BN_EPS 1e-3f
#define NB   32          // batch
#define HH   64
#define WW   64
#define NPIX 4096        // HH*WW
#define CIN  256
#define HDS  4           // heads
#define KD   16          // dim_k
#define VD   64          // dim_v

__device__ __forceinline__ v8f wmma_f32(v2f a, v2f b, v8f c) {
    // D = A(16x4 f32) * B(4x16 f32) + C(16x16 f32)
    return __builtin_amdgcn_wmma_f32_16x16x4_f32(
        /*neg_a=*/false, a, /*neg_b=*/false, b,
        /*c_mod=*/(short)0, c, /*reuse_a=*/false, /*reuse_b=*/false);
}

// ---------------------------------------------------------------------------
// Kernel 1: fused q/k/v projection GEMM with BN on q and v.
// One wave per 16-row tile of X[131072 x 256]; each wave produces all 144
// output columns (4 q-tiles, 1 k-tile, 4 v-tiles) so A is loaded once/step.
// ---------------------------------------------------------------------------
__global__ __launch_bounds__(256) void qkv_gemm(
    const float* __restrict__ X,  const float* __restrict__ Wq,
    const float* __restrict__ Wk, const float* __restrict__ Wv,
    const float* __restrict__ gq, const float* __restrict__ bq,
    const float* __restrict__ mq, const float* __restrict__ vq,
    const float* __restrict__ gv, const float* __restrict__ bv,
    const float* __restrict__ mv, const float* __restrict__ vv,
    float* __restrict__ qbuf, float* __restrict__ kbuf, float* __restrict__ vbuf)
{
    const int lane = threadIdx.x & 31;
    const int wave = threadIdx.x >> 5;
    const int r    = blockIdx.x * 8 + wave;   // row tile 0..8191
    const int col  = lane & 15;               // M for A-frag, N for B/C-frag
    const int half = lane >> 4;

    const float* xrow = X + (size_t)(r * 16 + col) * CIN;

    v8f zero = {0.f,0.f,0.f,0.f,0.f,0.f,0.f,0.f};
    v8f acc[9];
    #pragma unroll
    for (int t = 0; t < 9; ++t) acc[t] = zero;

    for (int kk = 0; kk < CIN; kk += 4) {
        const int k0 = kk + 2 * half;
        v2f a;                                 // A[M][k0], A[M][k0+1]
        a.x = xrow[k0];
        a.y = xrow[k0 + 1];
        #pragma unroll
        for (int t = 0; t < 4; ++t) {          // q tiles (ld=64)
            v2f bb;
            bb.x = Wq[(size_t)k0 * 64 + t * 16 + col];
            bb.y = Wq[(size_t)(k0 + 1) * 64 + t * 16 + col];
            acc[t] = wmma_f32(a, bb, acc[t]);
        }
        {                                      // k tile (ld=16)
            v2f bb;
            bb.x = Wk[(size_t)k0 * 16 + col];
            bb.y = Wk[(size_t)(k0 + 1) * 16 + col];
            acc[4] = wmma_f32(a, bb, acc[4]);
        }
        #pragma unroll
        for (int t = 0; t < 4; ++t) {          // v tiles (ld=64)
            v2f bb;
            bb.x = Wv[(size_t)k0 * 64 + t * 16 + col];
            bb.y = Wv[(size_t)(k0 + 1) * 64 + t * 16 + col];
            acc[5 + t] = wmma_f32(a, bb, acc[5 + t]);
        }
    }

    // C/D layout: reg i -> M = i + 8*half ; N = col
    #pragma unroll
    for (int t = 0; t < 4; ++t) {              // q with BN
        const int ch = t * 16 + col;
        const float sc = gq[ch] * rsqrtf(vq[ch] + BN_EPS);
        const float sh = bq[ch] - sc * mq[ch];
        #pragma unroll
        for (int i = 0; i < 8; ++i) {
            const int row = r * 16 + i + 8 * half;
            qbuf[(size_t)row * 64 + ch] = sc * acc[t][i] + sh;
        }
    }
    #pragma unroll
    for (int i = 0; i < 8; ++i) {              // k raw
        const int row = r * 16 + i + 8 * half;
        kbuf[(size_t)row * 16 + col] = acc[4][i];
    }
    #pragma unroll
    for (int t = 0; t < 4; ++t) {              // v with BN
        const int ch = t * 16 + col;
        const float sc = gv[ch] * rsqrtf(vv[ch] + BN_EPS);
        const float sh = bv[ch] - sc * mv[ch];
        #pragma unroll
        for (int i = 0; i < 8; ++i) {
            const int row = r * 16 + i + 8 * half;
            vbuf[(size_t)row * 64 + ch] = sc * acc[5 + t][i] + sh;
        }
    }
}

// ---------------------------------------------------------------------------
// Kernel 2a: per-(b, k-channel) softmax stats over 4096 positions.
// One wave per reduction; wave32 shuffle tree.
// ---------------------------------------------------------------------------
__global__ __launch_bounds__(256) void softmax_stats(
    const float* __restrict__ kbuf, float* __restrict__ smax, float* __restrict__ sinv)
{
    const int lane = threadIdx.x & 31;
    const int wave = threadIdx.x >> 5;
    const int w    = blockIdx.x * 8 + wave;    // 0..511
    const int b    = w >> 4;
    const int ch   = w & 15;
    const float* base = kbuf + (size_t)b * NPIX * KD + ch;

    float mx = -3.4e38f;
    for (int m = lane; m < NPIX; m += 32)
        mx = fmaxf(mx, base[(size_t)m * KD]);
    #pragma unroll
    for (int off = 16; off > 0; off >>= 1)
        mx = fmaxf(mx, __shfl_xor(mx, off, 32));

    float s = 0.f;
    for (int m = lane; m < NPIX; m += 32)
        s += expf(base[(size_t)m * KD] - mx);
    #pragma unroll
    for (int off = 16; off > 0; off >>= 1)
        s += __shfl_xor(s, off, 32);

    if (lane == 0) { smax[w] = mx; sinv[w] = 1.0f / s; }
}

__global__ void zero_lc(float* __restrict__ lc, int n)
{
    const int i = blockIdx.x * blockDim.x + threadIdx.x;
    if (i < n) lc[i] = 0.f;
}

// ---------------------------------------------------------------------------
// Kernel 2b: Lc[b][k][v] = sum_m softmax(k)[b,m,k] * v[b,m,v] via WMMA.
// m split 16 ways per batch; partials combined with global f32 atomics.
// ---------------------------------------------------------------------------
__global__ __launch_bounds__(256) void lc_gemm(
    const float* __restrict__ kbuf, const float* __restrict__ vbuf,
    const float* __restrict__ smax, const float* __restrict__ sinv,
    float* __restrict__ lc)
{
    const int lane  = threadIdx.x & 31;
    const int wave  = threadIdx.x >> 5;
    const int w     = blockIdx.x * 8 + wave;   // 0..511
    const int b     = w >> 4;
    const int chunk = w & 15;
    const int m0    = chunk * 256;
    const int col   = lane & 15;               // kchan for A, vcol for B
    const int half  = lane >> 4;

    const float mx  = smax[b * 16 + col];
    const float inv = sinv[b * 16 + col];
    const float* kb = kbuf + (size_t)b * NPIX * KD;
    const float* vb = vbuf + (size_t)b * NPIX * VD;

    v8f zero = {0.f,0.f,0.f,0.f,0.f,0.f,0.f,0.f};
    v8f acc[4] = {zero, zero, zero, zero};

    for (int mm = m0; mm < m0 + 256; mm += 4) {
        const int mA = mm + 2 * half;
        v2f a;                                 // A[kchan][m] = normalized softmax
        a.x = expf(kb[(size_t)mA * KD + col] - mx) * inv;
        a.y = expf(kb[(size_t)(mA + 1) * KD + col] - mx) * inv;
        #pragma unroll
        for (int t = 0; t < 4; ++t) {
            v2f bb;                            // B[m][vcol]
            bb.x = vb[(size_t)mA * VD + t * 16 + col];
            bb.y = vb[(size_t)(mA + 1) * VD + t * 16 + col];
            acc[t] = wmma_f32(a, bb, acc[t]);
        }
    }

    float* lcb = lc + (size_t)b * KD * VD;
    #pragma unroll
    for (int t = 0; t < 4; ++t)
        #pragma unroll
        for (int i = 0; i < 8; ++i) {
            const int krow = i + 8 * half;
            atomicAdd(&lcb[krow * VD + t * 16 + col], acc[t][i]);
        }
}

// ---------------------------------------------------------------------------
// Kernel 3: fused output. Per 8x8 pixel tile: stage 14x14x64 v-neighborhood,
// 7x7x16 filter and Lc in LDS. Yp uses the factored form
//   Yp[h,v] = sum_tap (sum_k q[h,k] w[tap,k]) * v[neighbor(tap), v]
// plus Yc = q.Lc and the pos_b term.
// ---------------------------------------------------------------------------
__global__ __launch_bounds__(64) void out_fused(
    const float* __restrict__ qbuf, const float* __restrict__ vbuf,
    const float* __restrict__ lc,   const float* __restrict__ posw,
    const float* __restrict__ posb, float* __restrict__ out)
{
    __shared__ float vlds[196 * 64];   // 14x14 neighborhood x 64 v-channels
    __shared__ float wlds[49 * 16];    // 7x7 taps x 16 k-channels
    __shared__ float lclds[16 * 64];   // Lc for this batch

    const int b  = blockIdx.z;
    const int y0 = blockIdx.y * 8;
    const int x0 = blockIdx.x * 8;
    const int tid = threadIdx.x;

    for (int i = tid; i < 49 * 16; i += 64) wlds[i] = posw[i];
    for (int i = tid; i < 16 * 64; i += 64) lclds[i] = lc[(size_t)b * KD * VD + i];
    for (int i = tid; i < 196 * 64; i += 64) {
        const int row = i >> 6, ch = i & 63;
        const int gy = y0 - 3 + row / 14;
        const int gx = x0 - 3 + row % 14;
        float val = 0.f;
        if (gy >= 0 && gy < HH && gx >= 0 && gx < WW)
            val = vbuf[(((size_t)b * HH + gy) * WW + gx) * VD + ch];
        vlds[i] = val;
    }
    __syncthreads();

    const int px = tid & 7, py = tid >> 3;
    const int p  = (y0 + py) * WW + (x0 + px);
    const float* qrow = qbuf + ((size_t)b * NPIX + p) * 64;
    float q[64];
    #pragma unroll
    for (int i = 0; i < 64; ++i) q[i] = qrow[i];

    float th[HDS];
    #pragma unroll
    for (int h = 0; h < HDS; ++h) {
        float s = 0.f;
        #pragma unroll
        for (int k = 0; k < KD; ++k) s += q[h * 16 + k] * posb[k];
        th[h] = s;
    }

    float* orow = out + ((size_t)b * NPIX + p) * 256;
    for (int vc = 0; vc < 4; ++vc) {           // 16 v-channels per chunk
        float acc[HDS][16];
        #pragma unroll
        for (int h = 0; h < HDS; ++h)
            #pragma unroll
            for (int j = 0; j < 16; ++j) acc[h][j] = th[h];

        #pragma unroll
        for (int k = 0; k < KD; ++k) {         // content lambda
            const float* lcr = &lclds[k * 64 + vc * 16];
            #pragma unroll
            for (int h = 0; h < HDS; ++h) {
                const float qq = q[h * 16 + k];
                #pragma unroll
                for (int j = 0; j < 16; ++j) acc[h][j] += qq * lcr[j];
            }
        }

        for (int dy = 0; dy < 7; ++dy)         // positional lambda
            for (int dx = 0; dx < 7; ++dx) {
                const float* wt = &wlds[(dy * 7 + dx) * 16];
                float s[HDS];
                #pragma unroll
                for (int h = 0; h < HDS; ++h) {
                    float a2 = 0.f;
                    #pragma unroll
                    for (int k = 0; k < KD; ++k) a2 += q[h * 16 + k] * wt[k];
                    s[h] = a2;
                }
                const float* vr = &vlds[((py + dy) * 14 + (px + dx)) * 64 + vc * 16];
                #pragma unroll
                for (int h = 0; h < HDS; ++h)
                    #pragma unroll
                    for (int j = 0; j < 16; ++j) acc[h][j] += s[h] * vr[j];
            }

        #pragma unroll
        for (int h = 0; h < HDS; ++h)
            #pragma unroll
            for (int j = 0; j < 16; ++j)
                orow[h * 64 + vc * 16 + j] = acc[h][j];
    }
}

// ---------------------------------------------------------------------------
extern "C" void kernel_launch(void* const* d_in, const int* in_sizes, int n_in,
                              void* d_out, int out_size, void* d_ws, size_t ws_size,
                              hipStream_t stream)
{
    (void)in_sizes; (void)n_in; (void)out_size; (void)ws_size;
    const float* X  = (const float*)d_in[0];
    const float* Wq = (const float*)d_in[1];
    const float* Wk = (const float*)d_in[2];
    const float* Wv = (const float*)d_in[3];
    const float* gq = (const float*)d_in[4];
    const float* bq = (const float*)d_in[5];
    const float* mq = (const float*)d_in[6];
    const float* vq = (const float*)d_in[7];
    const float* gv = (const float*)d_in[8];
    const float* bv = (const float*)d_in[9];
    const float* mv = (const float*)d_in[10];
    const float* vv = (const float*)d_in[11];
    const float* pw = (const float*)d_in[12];
    const float* pb = (const float*)d_in[13];
    float* out = (float*)d_out;

    float* ws   = (float*)d_ws;
    float* qbuf = ws;                                  // 131072*64
    float* kbuf = qbuf + (size_t)131072 * 64;          // 131072*16
    float* vbuf = kbuf + (size_t)131072 * 16;          // 131072*64
    float* smax = vbuf + (size_t)131072 * 64;          // 512
    float* sinv = smax + 512;                          // 512
    float* lcb  = sinv + 512;                          // 32*16*64

    qkv_gemm<<<1024, 256, 0, stream>>>(X, Wq, Wk, Wv,
                                       gq, bq, mq, vq, gv, bv, mv, vv,
                                       qbuf, kbuf, vbuf);
    softmax_stats<<<64, 256, 0, stream>>>(kbuf, smax, sinv);
    zero_lc<<<(NB * KD * VD + 255) / 256, 256, 0, stream>>>(lcb, NB * KD * VD);
    lc_gemm<<<64, 256, 0, stream>>>(kbuf, vbuf, smax, sinv, lcb);
    dim3 g3(8, 8, NB);
    out_fused<<<g3, 64, 0, stream>>>(qbuf, vbuf, lcb, pw, pb, out);
}